// weight_quantize_fn_17437567221967
// MI455X (gfx1250) — compile-verified
//
#include <hip/hip_runtime.h>
#include <math.h>

typedef __attribute__((ext_vector_type(2))) float v2f;
typedef __attribute__((ext_vector_type(4))) float v4f;
typedef __attribute__((ext_vector_type(8))) float v8f;

#define BLK 256

__device__ __forceinline__ float hsum8(v8f v) {
    float a = v[0] + v[1];
    float b = v[2] + v[3];
    float c = v[4] + v[5];
    float d = v[6] + v[7];
    return (a + b) + (c + d);
}

// ---------------------------------------------------------------------------
// Pass 1: block partial sums of x and x^2 via V_WMMA_F32_16X16X4_F32.
// A = all-ones 16x4, so D[m][n] = sum_k B[k][n] + C[m][n]: every WMMA folds 64
// values (2 VGPRs/lane) into the accumulator; every accumulator row holds the
// same column sums, so wave total = (sum over 8 VGPRs x 32 lanes) / 16.
// Loads use default RT temporal hint to deliberately warm the 192MB L2 with
// the 64MB weight for the second pass.
// ---------------------------------------------------------------------------
__global__ void stats_pass(const float* __restrict__ w, float* __restrict__ ws, int n) {
    const int tid = threadIdx.x;
    const long long gtid = (long long)blockIdx.x * BLK + tid;
    const long long tot  = (long long)gridDim.x * BLK;
    const long long n4   = (long long)n >> 2;        // # of full float4s
    const long long n4m  = n4 & ~31LL;               // multiple of 32 -> EXEC all-ones in loop

    v8f acc_s = {};
    v8f acc_q = {};
    v2f ones; ones[0] = 1.0f; ones[1] = 1.0f;
    float ts = 0.0f, tq = 0.0f;                      // scalar tail accumulators

    // Main loop: wave-first index (gtid - lane) is always a multiple of 32, and
    // n4m is a multiple of 32, so (i < n4m) is uniform across each wave.
    for (long long i = gtid; i < n4m; i += tot) {
        v4f v = *reinterpret_cast<const v4f*>(w + i * 4);
        v2f b0; b0[0] = v[0];        b0[1] = v[1];
        v2f b1; b1[0] = v[2];        b1[1] = v[3];
        v2f q0; q0[0] = v[0] * v[0]; q0[1] = v[1] * v[1];
        v2f q1; q1[0] = v[2] * v[2]; q1[1] = v[3] * v[3];
        acc_s = __builtin_amdgcn_wmma_f32_16x16x4_f32(false, ones, false, b0, (short)0, acc_s, false, false);
        acc_s = __builtin_amdgcn_wmma_f32_16x16x4_f32(false, ones, false, b1, (short)0, acc_s, false, false);
        acc_q = __builtin_amdgcn_wmma_f32_16x16x4_f32(false, ones, false, q0, (short)0, acc_q, false, false);
        acc_q = __builtin_amdgcn_wmma_f32_16x16x4_f32(false, ones, false, q1, (short)0, acc_q, false, false);
    }

    // Scalar tail (elements not covered by full 32-lane float4 groups).
    for (long long j = n4m * 4 + gtid; j < (long long)n; j += tot) {
        float x = w[j];
        ts += x;
        tq += x * x;
    }

    // Each WMMA accumulator counts every element 16x (16 identical rows).
    float local_s = hsum8(acc_s) * 0.0625f + ts;
    float local_q = hsum8(acc_q) * 0.0625f + tq;

    __shared__ float ls[BLK];
    __shared__ float lq[BLK];
    ls[tid] = local_s;
    lq[tid] = local_q;
    __syncthreads();
    for (int off = BLK / 2; off > 0; off >>= 1) {
        if (tid < off) {
            ls[tid] += ls[tid + off];
            lq[tid] += lq[tid + off];
        }
        __syncthreads();
    }
    if (tid == 0) {
        ws[blockIdx.x] = ls[0];
        ws[gridDim.x + blockIdx.x] = lq[0];
    }
}

// ---------------------------------------------------------------------------
// Pass 2: single-block finalize (f64): mean, unbiased std (ddof=1), and the
// three fused constants the quantize pass needs.
// ---------------------------------------------------------------------------
__global__ void finalize_stats(const float* __restrict__ ws_in, float* __restrict__ consts,
                               const float* __restrict__ alpha_p, int nb, long long n) {
    __shared__ double ds[BLK];
    __shared__ double dq[BLK];
    double s = 0.0, q = 0.0;
    for (int i = threadIdx.x; i < nb; i += BLK) {
        s += (double)ws_in[i];
        q += (double)ws_in[nb + i];
    }
    ds[threadIdx.x] = s;
    dq[threadIdx.x] = q;
    __syncthreads();
    for (int off = BLK / 2; off > 0; off >>= 1) {
        if (threadIdx.x < off) {
            ds[threadIdx.x] += ds[threadIdx.x + off];
            dq[threadIdx.x] += dq[threadIdx.x + off];
        }
        __syncthreads();
    }
    if (threadIdx.x == 0) {
        double sum = ds[0], sq = dq[0];
        double mean = sum / (double)n;
        double var  = (sq - sum * sum / (double)n) / (double)(n - 1);
        double sd   = sqrt(var);
        float alpha = alpha_p[0];
        consts[0] = (float)mean;                       // mean
        consts[1] = (float)(1.0 / (sd * (double)alpha)); // 1/(std*alpha)
        consts[2] = alpha * (1.0f / 15.0f);            // alpha/15 (grid step * alpha)
    }
}

// ---------------------------------------------------------------------------
// Pass 3: streaming quantize. Weight re-read hits L2 (warmed by pass 1);
// output written with non-temporal b128 stores so the write stream does not
// evict the cached weight. global_prefetch_b8 issued one grid-stride ahead.
// out = sign(t) * rint(min(|t|,1)*15) * (alpha/15),  t = (x-mean)/(std*alpha)
// ---------------------------------------------------------------------------
__global__ void quantize_pass(const float* __restrict__ w, float* __restrict__ out,
                              const float* __restrict__ consts, int n) {
    const float mean   = consts[0];
    const float s      = consts[1];
    const float oscale = consts[2];
    const long long gtid = (long long)blockIdx.x * blockDim.x + threadIdx.x;
    const long long tot  = (long long)gridDim.x * blockDim.x;
    const long long n4   = (long long)n >> 2;

    for (long long i = gtid; i < n4; i += tot) {
        __builtin_prefetch(w + (i + tot) * 4, 0, 0);   // speculative; OOB is dropped
        v4f v = *reinterpret_cast<const v4f*>(w + i * 4);
        v4f r;
#pragma unroll
        for (int k = 0; k < 4; ++k) {
            float t   = (v[k] - mean) * s;
            float a   = fminf(fabsf(t), 1.0f);
            float lvl = rintf(a * 15.0f);
            r[k] = copysignf(lvl * oscale, t);
        }
        __builtin_nontemporal_store(r, reinterpret_cast<v4f*>(out + i * 4));
    }
    // scalar tail
    for (long long j = n4 * 4 + gtid; j < (long long)n; j += tot) {
        float t   = (w[j] - mean) * s;
        float a   = fminf(fabsf(t), 1.0f);
        float lvl = rintf(a * 15.0f);
        out[j] = copysignf(lvl * oscale, t);
    }
}

extern "C" void kernel_launch(void* const* d_in, const int* in_sizes, int n_in,
                              void* d_out, int out_size, void* d_ws, size_t ws_size,
                              hipStream_t stream) {
    const float* w     = (const float*)d_in[0];
    const float* alpha = (const float*)d_in[1];
    float* out = (float*)d_out;
    float* ws  = (float*)d_ws;
    const int n = in_sizes[0];

    // Reduction block count: 2048 partials (16KB ws) unless scratch is tiny.
    int nb = 2048;
    size_t need = (size_t)(2 * nb + 4) * sizeof(float);
    if (ws_size < need) {
        long long avail = (long long)(ws_size / sizeof(float)) - 4;
        nb = (int)(avail / 2);
        if (nb < 1) nb = 1;
    }
    float* consts = ws + 2 * nb;

    stats_pass<<<nb, BLK, 0, stream>>>(w, ws, n);
    finalize_stats<<<1, BLK, 0, stream>>>(ws, consts, alpha, nb, (long long)n);

    // 4096 blocks x 256 threads x 4 floats/thread/iter -> 4 grid-stride iters for 16.7M.
    quantize_pass<<<4096, BLK, 0, stream>>>(w, out, consts, n);
}